// Net_11390253269710
// MI455X (gfx1250) — compile-verified
//
#include <hip/hip_runtime.h>

typedef __attribute__((ext_vector_type(2))) float v2f;
typedef __attribute__((ext_vector_type(8))) float v8f;

// ---------------------------------------------------------------- utilities
__global__ void zero_kernel(float* __restrict__ p, int n) {
  int i = blockIdx.x * blockDim.x + threadIdx.x;
  if (i < n) p[i] = 0.0f;
}

__device__ __forceinline__ void atomAddF(float* p, float v) {
#if defined(__HIP_DEVICE_COMPILE__)
  unsafeAtomicAdd(p, v);            // hardware global_atomic_add_f32
#else
  (void)p; (void)v;
#endif
}

// ----------------------------------------------------- degree / rsqrt(norm)
__global__ void deg_count_kernel(const int* __restrict__ dst, float* __restrict__ deg, int E) {
  int e = blockIdx.x * blockDim.x + threadIdx.x;
  if (e < E) atomAddF(&deg[dst[e]], 1.0f);
}

__global__ void dis_finalize_kernel(float* __restrict__ dis, int n) {
  int i = blockIdx.x * blockDim.x + threadIdx.x;
  if (i < n) {
    float d = dis[i] + 1.0f;                  // + self loop
    dis[i] = (d > 0.0f) ? rsqrtf(d) : 0.0f;   // deg^{-1/2}
  }
}

// ------------------------------------- layer 1: hs = (x[N,3] @ W[3,32]) * dis
// wave per node, lane = output feature
__global__ void gemm3_kernel(const float* __restrict__ x, const float* __restrict__ W,
                             const float* __restrict__ dis, float* __restrict__ hs,
                             float* __restrict__ agg, int n) {
  int lane = threadIdx.x & 31;
  int node = blockIdx.x * (blockDim.x >> 5) + (threadIdx.x >> 5);
  if (node >= n) return;
  float x0 = x[node * 3 + 0], x1 = x[node * 3 + 1], x2 = x[node * 3 + 2];
  float acc = fmaf(x0, W[lane], fmaf(x1, W[32 + lane], x2 * W[64 + lane]));
  float v = acc * dis[node];
  hs[node * 32 + lane]  = v;
  agg[node * 32 + lane] = v;   // self-loop contribution
}

// ---------------------- layer 2: hs = (X[N,32] @ W[32,32]) * dis  via WMMA f32
// One wave handles a 16-row tile. K=32 in 8 steps of 16x16x4, two 16-col tiles.
// A (16x4 f32): lanes 0-15 hold K={k0,k0+1}, lanes 16-31 hold K={k0+2,k0+3}.
// C/D (16x16 f32): VGPR v, lanes 0-15 -> M=v, lanes 16-31 -> M=8+v.
__global__ void gemm32_wmma_kernel(const float* __restrict__ X, const float* __restrict__ W,
                                   const float* __restrict__ dis, float* __restrict__ hs,
                                   float* __restrict__ agg, int nTiles) {
  int lane    = threadIdx.x & 31;
  int tile    = blockIdx.x * (blockDim.x >> 5) + (threadIdx.x >> 5);
  if (tile >= nTiles) return;              // wave-uniform: EXEC stays all-ones
  int halfSel = lane >> 4;                 // 0: K pair 0/1, 1: K pair 2/3
  int lane16  = lane & 15;
  int rowBase = tile * 16;

  const float* Xrow = X + (size_t)(rowBase + lane16) * 32;
  v8f c0 = {}; v8f c1 = {};
#pragma unroll
  for (int ks = 0; ks < 8; ++ks) {
    int ka = ks * 4 + 2 * halfSel;
    v2f a = *(const v2f*)(Xrow + ka);                       // A[m][ka], A[m][ka+1]
    v2f b0, b1;
    b0.x = W[ka * 32 + lane16];       b0.y = W[(ka + 1) * 32 + lane16];
    b1.x = W[ka * 32 + 16 + lane16];  b1.y = W[(ka + 1) * 32 + 16 + lane16];
    c0 = __builtin_amdgcn_wmma_f32_16x16x4_f32(false, a, false, b0, (short)0, c0, false, false);
    c1 = __builtin_amdgcn_wmma_f32_16x16x4_f32(false, a, false, b1, (short)0, c1, false, false);
  }
#pragma unroll
  for (int v = 0; v < 8; ++v) {
    int row = rowBase + v + 8 * halfSel;
    float dsc = dis[row];
    float v0 = c0[v] * dsc;
    float v1 = c1[v] * dsc;
    hs[row * 32 + lane16]       = v0;  agg[row * 32 + lane16]      = v0;
    hs[row * 32 + 16 + lane16]  = v1;  agg[row * 32 + 16 + lane16] = v1;
  }
}

// remainder rows (n % 16) — plain VALU, wave per node (not used when n%16==0)
__global__ void gemm32_rem_kernel(const float* __restrict__ x, const float* __restrict__ W,
                                  const float* __restrict__ dis, float* __restrict__ hs,
                                  float* __restrict__ agg, int start, int n) {
  int lane = threadIdx.x & 31;
  int node = start + blockIdx.x * (blockDim.x >> 5) + (threadIdx.x >> 5);
  if (node >= n) return;
  float acc = 0.0f;
#pragma unroll
  for (int k = 0; k < 32; ++k) acc = fmaf(x[node * 32 + k], W[k * 32 + lane], acc);
  float v = acc * dis[node];
  hs[node * 32 + lane]  = v;
  agg[node * 32 + lane] = v;
}

// --------------------- layer 3: hs = (X[N,32] @ W[32,1]) * dis, wave reduction
__global__ void gemm1_kernel(const float* __restrict__ x, const float* __restrict__ W,
                             const float* __restrict__ dis, float* __restrict__ hs,
                             float* __restrict__ agg, int n) {
  int lane = threadIdx.x & 31;
  int node = blockIdx.x * (blockDim.x >> 5) + (threadIdx.x >> 5);
  if (node >= n) return;
  float p = x[node * 32 + lane] * W[lane];
#pragma unroll
  for (int off = 16; off > 0; off >>= 1) p += __shfl_xor(p, off, 32);
  if (lane == 0) {
    float v = p * dis[node];
    hs[node]  = v;
    agg[node] = v;
  }
}

// ------------------------------------------------ edge scatter, F=32 features
// one wave per edge; lane = feature -> 128B coalesced gather + 128B atomic burst
__global__ void scatter32_kernel(const int* __restrict__ src, const int* __restrict__ dst,
                                 const float* __restrict__ hs, float* __restrict__ agg, int E) {
  int lane = threadIdx.x & 31;
  int e = blockIdx.x * (blockDim.x >> 5) + (threadIdx.x >> 5);
  if (e >= E) return;
  __builtin_prefetch(src + e + 2048, 0, 1);   // stream the edge list (global_prefetch_b8)
  __builtin_prefetch(dst + e + 2048, 0, 1);
  int s = src[e];                              // wave-uniform
  int d = dst[e];
  float v = hs[(size_t)s * 32 + lane];
  atomAddF(&agg[(size_t)d * 32 + lane], v);
}

// edge scatter, F=1
__global__ void scatter1_kernel(const int* __restrict__ src, const int* __restrict__ dst,
                                const float* __restrict__ hs, float* __restrict__ agg, int E) {
  int e = blockIdx.x * blockDim.x + threadIdx.x;
  if (e < E) atomAddF(&agg[dst[e]], hs[src[e]]);
}

// ------------------------------------------------- out = act(dis*agg + b)
__global__ void finalize32_kernel(const float* __restrict__ agg, const float* __restrict__ dis,
                                  const float* __restrict__ b, float* __restrict__ out,
                                  int n, int do_relu) {
  int i = blockIdx.x * blockDim.x + threadIdx.x;
  if (i >= n * 32) return;
  int node = i >> 5, f = i & 31;
  float v = fmaf(dis[node], agg[i], b[f]);
  out[i] = do_relu ? fmaxf(v, 0.0f) : v;
}

__global__ void finalize1_kernel(const float* __restrict__ agg, const float* __restrict__ dis,
                                 const float* __restrict__ b, float* __restrict__ out, int n) {
  int i = blockIdx.x * blockDim.x + threadIdx.x;
  if (i < n) out[i] = fmaf(dis[i], agg[i], b[0]);
}

// ---------------------------------------------------------------- launcher
extern "C" void kernel_launch(void* const* d_in, const int* in_sizes, int n_in,
                              void* d_out, int out_size, void* d_ws, size_t ws_size,
                              hipStream_t stream) {
  const float* x  = (const float*)d_in[0];
  const int*   ei = (const int*)  d_in[1];
  const float* W1 = (const float*)d_in[2];
  const float* b1 = (const float*)d_in[3];
  const float* W2 = (const float*)d_in[4];
  const float* b2 = (const float*)d_in[5];
  const float* W3 = (const float*)d_in[6];
  const float* b3 = (const float*)d_in[7];
  float* out = (float*)d_out;

  const int n = in_sizes[0] / 3;
  const int E = in_sizes[1] / 2;
  const int* src = ei;
  const int* dst = ei + E;

  float* ws  = (float*)d_ws;
  float* dis = ws;                        // [n]
  float* x1  = dis + n;                   // [n*32]
  float* x2  = x1 + (size_t)n * 32;       // [n*32]
  float* hs  = x2 + (size_t)n * 32;       // [n*32]  (layer3 reuses first n)
  float* agg = hs + (size_t)n * 32;       // [n*32]  (layer3 reuses first n)

  const int B = 256;                      // 8 wave32s per block
  const int WPB = B / 32;
  auto cdiv = [](int a, int b) { return (a + b - 1) / b; };

  // ---- symmetric normalization dis = (deg+1)^{-1/2}
  zero_kernel        <<<cdiv(n, B), B, 0, stream>>>(dis, n);
  deg_count_kernel   <<<cdiv(E, B), B, 0, stream>>>(dst, dis, E);
  dis_finalize_kernel<<<cdiv(n, B), B, 0, stream>>>(dis, n);

  // ---- layer 1: [N,3] -> [N,32], relu
  gemm3_kernel    <<<cdiv(n, WPB),      B, 0, stream>>>(x, W1, dis, hs, agg, n);
  scatter32_kernel<<<cdiv(E, WPB),      B, 0, stream>>>(src, dst, hs, agg, E);
  finalize32_kernel<<<cdiv(n * 32, B),  B, 0, stream>>>(agg, dis, b1, x1, n, 1);

  // ---- layer 2: [N,32] -> [N,32] via V_WMMA_F32_16X16X4_F32, relu
  int nFullTiles = n / 16;
  if (nFullTiles > 0)
    gemm32_wmma_kernel<<<cdiv(nFullTiles, WPB), B, 0, stream>>>(x1, W2, dis, hs, agg, nFullTiles);
  int remStart = nFullTiles * 16;
  if (remStart < n)
    gemm32_rem_kernel<<<cdiv(n - remStart, WPB), B, 0, stream>>>(x1, W2, dis, hs, agg, remStart, n);
  scatter32_kernel <<<cdiv(E, WPB),     B, 0, stream>>>(src, dst, hs, agg, E);
  finalize32_kernel<<<cdiv(n * 32, B),  B, 0, stream>>>(agg, dis, b2, x2, n, 1);

  // ---- layer 3: [N,32] -> [N,1]
  gemm1_kernel    <<<cdiv(n, WPB), B, 0, stream>>>(x2, W3, dis, hs, agg, n);
  scatter1_kernel <<<cdiv(E, B),   B, 0, stream>>>(src, dst, hs, agg, E);
  finalize1_kernel<<<cdiv(n, B),   B, 0, stream>>>(agg, dis, b3, out, n);
}